// SpikeXOR_76141180223608
// MI455X (gfx1250) — compile-verified
//
#include <hip/hip_runtime.h>
#include <hip/hip_bf16.h>
#include <math.h>
#include <stdint.h>

typedef float v2f __attribute__((ext_vector_type(2)));
typedef float v8f __attribute__((ext_vector_type(8)));

namespace {
constexpr int B_   = 128;
constexpr int T_   = 2048;
constexpr int H_   = 512;
constexpr int TC   = 128;        // timesteps staged per LDS chunk
constexpr int NCH  = T_ / TC;    // 16 chunks
constexpr int WV   = 4;          // h-tiles (waves) per block
constexpr int NHT  = H_ / 16;    // 32 h-tiles total
}

// Issue async global->LDS copies for one chunk: 16 b-rows x TC steps of 8B
// (x[b][t][0..1]) into interleaved layout xs[t][row][k] at (t*32+row)*8 bytes.
// Row m is copied by the 128 threads covering t=0..127 -> fully coalesced 1KB.
__device__ __forceinline__ void async_stage_chunk(const float* __restrict__ x,
                                                  int b_base, int t0,
                                                  float* __restrict__ buf,
                                                  int tid)
{
    const uint32_t lds_base = (uint32_t)(uintptr_t)buf;   // wave-relative LDS addr
#pragma unroll
    for (int m = 0; m < 16; ++m) {
        const float*  g   = x + ((size_t)(b_base + m) * T_ + t0 + tid) * 2;
        const uint32_t l  = lds_base + (uint32_t)((tid * 64 + m * 2) * 4);
        asm volatile("global_load_async_to_lds_b64 %0, %1, off"
                     :: "v"(l), "v"(g) : "memory");
    }
}

__device__ __forceinline__ void wait_async_all() {
    asm volatile("s_wait_asynccnt 0x0" ::: "memory");
}

// One wave owns a 16(b) x 16(h) tile. Per timestep:
//   cur = WMMA_f32_16x16x4( A=[x0,x1|0,0], B=[Win0,Win1|0,0], C=b_in )
//   v   = v*(1-d) + cur*d ; s = (v>=th) ; cnt += s ; v = s ? 0 : v
// A-lanes 16..31 (K=2,3) read pre-zeroed LDS rows -> no per-step masking.
// x chunks are double-buffered via GLOBAL_LOAD_ASYNC_TO_LDS (ASYNCcnt).
__global__ __launch_bounds__(128)
void snn_scan_wmma_kernel(const float* __restrict__ x,      // [B,T,2]
                          const float* __restrict__ W_in,   // [2,H]
                          const float* __restrict__ b_in,   // [H]
                          const float* __restrict__ w_tau,  // [H]
                          const float* __restrict__ thresh, // [H]
                          const float* __restrict__ W_out,  // [H,2]
                          float* __restrict__ ws)           // [B*2*NHT] partials
{
    // double-buffered: [buf][t][row(32)][k(2)] ; rows 16..31 stay zero (K=2,3 pad)
    __shared__ __align__(16) float xs[2][TC * 64];

    const int tid  = threadIdx.x;
    const int lane = tid & 31;
    const int wave = tid >> 5;
    const int n    = lane & 15;   // column within tile (h index)
    const int half = lane >> 4;   // 0: M rows 0..7 / K0,1 ; 1: M rows 8..15 / K2,3

    const int b_base = (blockIdx.x >> 3) * 16;
    const int h_tile = (blockIdx.x & 7) * WV + wave;
    const int h_base = h_tile * 16;
    const int h      = h_base + n;

    // Per-column constants (uniform across the 8 accumulator VGPRs)
    const float d   = 1.0f / (1.0f + expf(-w_tau[h]));   // sigmoid(w_tau)
    const float omd = 1.0f - d;
    const float th  = thresh[h];
    const float bi  = b_in[h];

    // B operand: rows K=0,1 valid, K=2,3 (lanes 16..31) zero.
    v2f bm;
    bm.x = half ? 0.0f : W_in[h];        // K=0 / K=2
    bm.y = half ? 0.0f : W_in[H_ + h];   // K=1 / K=3

    // C operand: bias broadcast over all 8 rows held by this lane.
    v8f cb;
#pragma unroll
    for (int r = 0; r < 8; ++r) cb[r] = bi;

    v8f v   = {};
    v8f cnt = {};   // float: spike counts <= 2048 are exact, enables v_pk_add_f32

    // Zero the K-padding rows (16..31) of both buffers once.
    for (int q = tid; q < TC * 16 * 2; q += 128) {
        const int tl = q >> 5;
        const int rk = q & 31;
        xs[0][tl * 64 + 32 + rk] = 0.0f;
        xs[1][tl * 64 + 32 + rk] = 0.0f;
    }

    // Prologue: bring in chunk 0, then sync (also covers the zero-init above).
    async_stage_chunk(x, b_base, 0, xs[0], tid);
    wait_async_all();
    __syncthreads();

    for (int ch = 0; ch < NCH; ++ch) {
        const float* __restrict__ buf = xs[ch & 1];

        // Kick off the next chunk's async copy; it lands while we compute.
        if (ch + 1 < NCH)
            async_stage_chunk(x, b_base, (ch + 1) * TC, xs[(ch + 1) & 1], tid);

#pragma unroll 4
        for (int tl = 0; tl < TC; ++tl) {
            // A operand: lanes 0..15 -> {x[b=lane][t][0], x[..][1]}, lanes 16..31 -> 0
            const float2 xv = *(const float2*)(&buf[tl * 64 + lane * 2]);
            v2f a; a.x = xv.x; a.y = xv.y;

            v8f cur = __builtin_amdgcn_wmma_f32_16x16x4_f32(
                false, a, false, bm, (short)0, cb, false, false);

#pragma unroll
            for (int r = 0; r < 8; ++r) {
                const float vv = v[r] * omd + cur[r] * d;   // leaky integrate
                const bool  s  = (vv >= th);                // Heaviside
                cnt[r] += s ? 1.0f : 0.0f;                  // spike count
                v[r]    = s ? 0.0f : vv;                    // hard reset
            }
        }

        // Our wave's async copies done; barrier makes all waves' copies visible
        // and guarantees everyone finished reading `buf` before it is reused.
        wait_async_all();
        __syncthreads();
    }

    // rate @ W_out partials for this tile: reduce the 16 columns per row.
    const float invT = 1.0f / (float)T_;
    const float wo0  = W_out[h * 2 + 0];
    const float wo1  = W_out[h * 2 + 1];

#pragma unroll
    for (int r = 0; r < 8; ++r) {
        const float rate = cnt[r] * invT;
        float p0 = rate * wo0;
        float p1 = rate * wo1;
#pragma unroll
        for (int mask = 8; mask >= 1; mask >>= 1) {   // butterfly within 16-lane halves
            p0 += __shfl_xor(p0, mask, 16);
            p1 += __shfl_xor(p1, mask, 16);
        }
        if (n == 0) {
            const int bg = b_base + r + 8 * half;     // global batch row
            ws[(bg * 2 + 0) * NHT + h_tile] = p0;
            ws[(bg * 2 + 1) * NHT + h_tile] = p1;
        }
    }
}

// Deterministic fixed-order reduction of the 32 h-tile partials + bias.
__global__ __launch_bounds__(256)
void snn_finish_kernel(const float* __restrict__ ws,
                       const float* __restrict__ b_out,
                       float* __restrict__ out)
{
    const int i = threadIdx.x;            // 0..255 -> (b, j)
    float s = b_out[i & 1];
    const float* p = ws + (size_t)i * NHT;
#pragma unroll
    for (int ht = 0; ht < NHT; ++ht) s += p[ht];
    out[i] = s;
}

extern "C" void kernel_launch(void* const* d_in, const int* in_sizes, int n_in,
                              void* d_out, int out_size, void* d_ws, size_t ws_size,
                              hipStream_t stream) {
    (void)in_sizes; (void)n_in; (void)out_size; (void)ws_size;
    const float* x      = (const float*)d_in[0];  // [128,2048,2]
    const float* W_in   = (const float*)d_in[1];  // [2,512]
    const float* b_in   = (const float*)d_in[2];  // [512]
    const float* w_tau  = (const float*)d_in[3];  // [512]
    const float* thresh = (const float*)d_in[4];  // [512]
    const float* W_out  = (const float*)d_in[5];  // [512,2]
    const float* b_out  = (const float*)d_in[6];  // [2]
    float*       ws     = (float*)d_ws;           // needs 128*2*32*4 = 32 KB
    float*       out    = (float*)d_out;          // [128,2]

    snn_scan_wmma_kernel<<<dim3(64), dim3(128), 0, stream>>>(
        x, W_in, b_in, w_tau, thresh, W_out, ws);
    snn_finish_kernel<<<dim3(1), dim3(256), 0, stream>>>(ws, b_out, out);
}